// GAT_2293512536203
// MI455X (gfx1250) — compile-verified
//
#include <hip/hip_runtime.h>

typedef __attribute__((ext_vector_type(2))) float v2f;
typedef __attribute__((ext_vector_type(8))) float v8f;

#define NSRC 768
#define NTGT 768
#define FEAT 33
#define NTOT (NSRC + NTGT)
#define NELT (NTOT * FEAT)
#define KELEMS (NSRC * NTGT)

#define SINK_WGS 48
#define SINK_ITERS 1000

// ---------------------------------------------------------------------------
// Device-wide sense-reversal barrier (persistent-kernel sync through L2).
// bar[0] = arrival count, bar[1] = generation. Reset by the init kernel each
// call, so every replay starts from (0,0) deterministically.
// ---------------------------------------------------------------------------
__device__ __forceinline__ void gbar(unsigned* bar, unsigned* localGen) {
    __syncthreads();
    if (threadIdx.x == 0) {
        unsigned target = ++(*localGen);
        __threadfence();  // make this WG's global writes visible device-wide
        unsigned prev = atomicAdd(bar, 1u);
        if (prev == (unsigned)(SINK_WGS - 1)) {
            __hip_atomic_store(bar, 0u, __ATOMIC_RELAXED, __HIP_MEMORY_SCOPE_AGENT);
            __threadfence();
            __hip_atomic_store(bar + 1, target, __ATOMIC_RELEASE, __HIP_MEMORY_SCOPE_AGENT);
        } else {
            while (__hip_atomic_load(bar + 1, __ATOMIC_ACQUIRE, __HIP_MEMORY_SCOPE_AGENT) < target) {
                __builtin_amdgcn_s_sleep(2);  // low-power spin
            }
        }
        __threadfence();
    }
    __syncthreads();
}

// ---------------------------------------------------------------------------
// Kernel 1: global min/max of x, init barrier state and c = 1.
// ---------------------------------------------------------------------------
__global__ void __launch_bounds__(1024) sk_minmax_init(
    const float* __restrict__ x, float* __restrict__ mnmx,
    float* __restrict__ cvec, unsigned* __restrict__ bar) {
    __shared__ float smn[1024];
    __shared__ float smx[1024];
    const int t = threadIdx.x;
    float mn = 3.402823466e38f, mx = -3.402823466e38f;
    for (int i = t; i < NELT; i += 1024) {
        float v = x[i];
        mn = fminf(mn, v);
        mx = fmaxf(mx, v);
    }
    smn[t] = mn; smx[t] = mx;
    __syncthreads();
    for (int s = 512; s > 0; s >>= 1) {
        if (t < s) {
            smn[t] = fminf(smn[t], smn[t + s]);
            smx[t] = fmaxf(smx[t], smx[t + s]);
        }
        __syncthreads();
    }
    if (t == 0) {
        mnmx[0] = smn[0];
        mnmx[1] = smx[0];
        bar[0] = 0u;   // barrier arrival count
        bar[1] = 0u;   // barrier generation
    }
    if (t < NTGT) cvec[t] = 1.0f;   // Sinkhorn column scaling starts at 1
}

// ---------------------------------------------------------------------------
// Kernel 2: K = exp( (srcn . tgtn^T) / sqrt(33) ) via V_WMMA_F32_16X16X4_F32.
// One 16x16 output tile per wave; 9 K-steps of 4 (k = 33..35 zero padded).
// A layout (ISA 7.12.2, 32-bit A 16x4): lane = M, half-wave selects K pair.
// ---------------------------------------------------------------------------
__global__ void __launch_bounds__(256) sk_scores_exp(
    const float* __restrict__ x, const float* __restrict__ mnmx,
    float* __restrict__ Kmat) {
    const int lane = threadIdx.x & 31;
    const int wave = threadIdx.x >> 5;
    const int tile = blockIdx.x * 8 + wave;     // 0 .. 2303
    const int tm = tile / (NTGT / 16);
    const int tn = tile % (NTGT / 16);
    const int m0 = tm * 16, n0 = tn * 16;
    const int half = lane >> 4;                 // 0: K pair {0,1}, 1: {2,3}
    const int lr = lane & 15;

    const float mn  = mnmx[0];
    const float inv = 1.0f / (mnmx[1] - mn);

    const float* __restrict__ srcp = x + (m0 + lr) * FEAT;           // src row m0+lr
    const float* __restrict__ tgtp = x + (NSRC + n0 + lr) * FEAT;    // tgt row n0+lr

    v8f acc = {};
    for (int s = 0; s < 9; ++s) {
        const int k0 = s * 4 + 2 * half;
        v2f a, b;
        a.x = (k0     < FEAT) ? (srcp[k0]     - mn) * inv : 0.0f;
        a.y = (k0 + 1 < FEAT) ? (srcp[k0 + 1] - mn) * inv : 0.0f;
        b.x = (k0     < FEAT) ? (tgtp[k0]     - mn) * inv : 0.0f;
        b.y = (k0 + 1 < FEAT) ? (tgtp[k0 + 1] - mn) * inv : 0.0f;
        // 8 args: (neg_a, A, neg_b, B, c_mod, C, reuse_a, reuse_b)
        acc = __builtin_amdgcn_wmma_f32_16x16x4_f32(
            false, a, false, b, (short)0, acc, false, false);
    }
    const float scale = 0.17407765595569785f;   // 1/sqrt(33)
#pragma unroll
    for (int v = 0; v < 8; ++v) {
        const int row = m0 + v + 8 * half;      // C/D layout: VGPR v, half-wave
        const int col = n0 + lr;
        Kmat[row * NTGT + col] = __expf(acc[v] * scale);
    }
}

// ---------------------------------------------------------------------------
// Kernel 3: persistent Sinkhorn in scaling-vector form.
//   r = 1 / (K c) ; c = 1 / (K^T r) ; repeat 1000x ; out = diag(r) K diag(c)
// 48 WGs x 256 threads; WG owns 16 rows (r) and 16 cols (c). K stays in L2.
// ---------------------------------------------------------------------------
__global__ void __launch_bounds__(256) sk_sinkhorn(
    const float* __restrict__ Kmat, float* __restrict__ r, float* __restrict__ c,
    unsigned* __restrict__ bar, float* __restrict__ out) {
    __shared__ float vec[NTGT];     // broadcast copy of c (or r)
    __shared__ float red[256];
    const int t   = threadIdx.x;
    const int wg  = blockIdx.x;
    const int sub = t & 15;         // 0..15
    const int grp = t >> 4;         // 0..15
    unsigned localGen = 0;

    for (int it = 0; it < SINK_ITERS; ++it) {
        // ---- r update: r_i = 1 / sum_j K[i][j] * c[j] ----
        for (int j = t; j < NTGT; j += 256) vec[j] = c[j];
        __syncthreads();
        {
            const int row = wg * 16 + grp;
            const float* __restrict__ kr = Kmat + row * NTGT;
            float sum = 0.0f;
            for (int jb = sub * 4; jb < NTGT; jb += 64) {   // b128 loads
                float4 k4 = *(const float4*)(kr + jb);
                float4 c4 = *(const float4*)(vec + jb);
                sum += k4.x * c4.x + k4.y * c4.y + k4.z * c4.z + k4.w * c4.w;
            }
            for (int m = 8; m >= 1; m >>= 1) sum += __shfl_xor(sum, m, 16);
            if (sub == 0) r[row] = 1.0f / sum;
        }
        gbar(bar, &localGen);

        // ---- c update: c_j = 1 / sum_i K[i][j] * r[i] ----
        for (int j = t; j < NSRC; j += 256) vec[j] = r[j];
        __syncthreads();
        {
            const int col = wg * 16 + sub;
            float sum = 0.0f;
            const int i0 = grp * 48;
            for (int i = i0; i < i0 + 48; ++i)
                sum += Kmat[i * NTGT + col] * vec[i];       // coalesced over sub
            red[t] = sum;
            __syncthreads();
            if (t < 16) {
                float tot = 0.0f;
#pragma unroll
                for (int k = 0; k < 16; ++k) tot += red[t + 16 * k];
                c[wg * 16 + t] = 1.0f / tot;
            }
        }
        gbar(bar, &localGen);
    }

    // ---- final output: P[i][j] = r[i] * K[i][j] * c[j] ----
    for (int j = t; j < NTGT; j += 256) vec[j] = c[j];
    __syncthreads();
    {
        const int row = wg * 16 + grp;
        const float ri = r[row];
        const float* __restrict__ kr = Kmat + row * NTGT;
        float* __restrict__ orow = out + row * NTGT;
        for (int jb = sub * 4; jb < NTGT; jb += 64) {
            float4 k4 = *(const float4*)(kr + jb);
            float4 c4 = *(const float4*)(vec + jb);
            float4 o;
            o.x = ri * k4.x * c4.x;
            o.y = ri * k4.y * c4.y;
            o.z = ri * k4.z * c4.z;
            o.w = ri * k4.w * c4.w;
            *(float4*)(orow + jb) = o;
        }
    }
}

// ---------------------------------------------------------------------------
extern "C" void kernel_launch(void* const* d_in, const int* in_sizes, int n_in,
                              void* d_out, int out_size, void* d_ws, size_t ws_size,
                              hipStream_t stream) {
    (void)in_sizes; (void)n_in; (void)out_size; (void)ws_size;
    const float* x = (const float*)d_in[0];
    float* out = (float*)d_out;

    // workspace layout
    float*    Kmat = (float*)d_ws;            // 768*768 floats = 2.25 MB
    float*    r    = Kmat + KELEMS;           // 768
    float*    c    = r + NSRC;                // 768
    float*    mnmx = c + NTGT;                // 2
    unsigned* bar  = (unsigned*)(mnmx + 2);   // 2 (count, gen)

    sk_minmax_init<<<1, 1024, 0, stream>>>(x, mnmx, c, bar);
    sk_scores_exp<<<(48 * 48) / 8, 256, 0, stream>>>(x, mnmx, Kmat);
    sk_sinkhorn<<<SINK_WGS, 256, 0, stream>>>(Kmat, r, c, bar, out);
}